// CAPMemory_44607530336551
// MI455X (gfx1250) — compile-verified
//
#include <hip/hip_runtime.h>
#include <math.h>

typedef __attribute__((ext_vector_type(2))) float v2f;
typedef __attribute__((ext_vector_type(4))) float f4;
typedef __attribute__((ext_vector_type(8))) float v8f;

#define BSZ   256
#define DD    2048
#define NCAM  8
#define CC    2048
#define MTOT  (NCAM * CC)      // 16384
#define BETA  0.05f
#define KTOP  51               // >= 50 (asso, excl g) and >= 8 (online, excl 3)
#define TOPC  3
#define EXTRA 5

#define MT    64               // block M tile (4 x 16 per wave)
#define KT    64               // K chunk

// =====================================================================
// Kernel 1: sims[256][16384] = features[256][2048] @ tempV[16384][2048]^T
// fp32 WMMA 16x16x4. Block = 256 thr (8 waves), tile M=64 x N=128,
// double-buffered LDS, 4 accumulators per wave sharing each B fragment.
// =====================================================================
__global__ __launch_bounds__(256)
void capmem_gemm(const float* __restrict__ A,   // features
                 const float* __restrict__ V,   // mem flattened [16384][2048]
                 float* __restrict__ S)         // sims
{
    __shared__ __align__(16) float As[2][MT][68];     // 34.8 KB
    __shared__ __align__(16) float Bs[2][128][68];    // 69.6 KB

    const int tid  = threadIdx.x;
    const int wave = tid >> 5;
    const int lane = tid & 31;
    const int n0   = blockIdx.x * 128;
    const int m0   = blockIdx.y * MT;

    v8f acc[4] = {{}, {}, {}, {}};

    const int mrow = lane & 15;
    const int nrow = (wave << 4) + (lane & 15);
    const int ksel = (lane >> 4) << 1;          // 0 for lanes 0-15, 2 for 16-31

    f4 areg[4];
    f4 breg[8];

    // ---- prologue: stage chunk 0 into buffer 0 ----
    #pragma unroll
    for (int j = 0; j < 4; ++j) {
        int flat = tid + 256 * j, row = flat >> 4, kk = (flat & 15) << 2;
        areg[j] = *(const f4*)(A + (size_t)(m0 + row) * DD + kk);
    }
    #pragma unroll
    for (int j = 0; j < 8; ++j) {
        int flat = tid + 256 * j, row = flat >> 4, kk = (flat & 15) << 2;
        breg[j] = *(const f4*)(V + (size_t)(n0 + row) * DD + kk);
    }
    #pragma unroll
    for (int j = 0; j < 4; ++j) {
        int flat = tid + 256 * j, row = flat >> 4, kk = (flat & 15) << 2;
        *(f4*)&As[0][row][kk] = areg[j];
    }
    #pragma unroll
    for (int j = 0; j < 8; ++j) {
        int flat = tid + 256 * j, row = flat >> 4, kk = (flat & 15) << 2;
        *(f4*)&Bs[0][row][kk] = breg[j];
    }
    __syncthreads();

    const int NCHUNK = DD / KT;   // 32
    for (int c = 0; c < NCHUNK; ++c) {
        const int buf = c & 1;
        const int k0n = (c + 1) * KT;

        // ---- prefetch next chunk into registers (overlaps with WMMA) ----
        if (c + 1 < NCHUNK) {
            #pragma unroll
            for (int j = 0; j < 4; ++j) {
                int flat = tid + 256 * j, row = flat >> 4, kk = (flat & 15) << 2;
                areg[j] = *(const f4*)(A + (size_t)(m0 + row) * DD + k0n + kk);
            }
            #pragma unroll
            for (int j = 0; j < 8; ++j) {
                int flat = tid + 256 * j, row = flat >> 4, kk = (flat & 15) << 2;
                breg[j] = *(const f4*)(V + (size_t)(n0 + row) * DD + k0n + kk);
            }
        }

        // ---- 64 WMMA ops on current buffer; B fragment shared by 4 tiles ----
        #pragma unroll
        for (int kk = 0; kk < KT; kk += 4) {
            v2f b;
            b.x = Bs[buf][nrow][kk + ksel];
            b.y = Bs[buf][nrow][kk + ksel + 1];
            #pragma unroll
            for (int m = 0; m < 4; ++m) {
                v2f a;
                a.x = As[buf][m * 16 + mrow][kk + ksel];
                a.y = As[buf][m * 16 + mrow][kk + ksel + 1];
                acc[m] = __builtin_amdgcn_wmma_f32_16x16x4_f32(
                    false, a, false, b, (short)0, acc[m], false, false);
            }
        }

        // ---- drain prefetch registers into the idle buffer ----
        if (c + 1 < NCHUNK) {
            #pragma unroll
            for (int j = 0; j < 4; ++j) {
                int flat = tid + 256 * j, row = flat >> 4, kk = (flat & 15) << 2;
                *(f4*)&As[buf ^ 1][row][kk] = areg[j];
            }
            #pragma unroll
            for (int j = 0; j < 8; ++j) {
                int flat = tid + 256 * j, row = flat >> 4, kk = (flat & 15) << 2;
                *(f4*)&Bs[buf ^ 1][row][kk] = breg[j];
            }
        }
        __syncthreads();
    }

    // ---- store C: VGPR r -> M = r (lanes 0-15) / 8+r (lanes 16-31), N = lane&15
    const int ncol = n0 + (wave << 4) + (lane & 15);
    #pragma unroll
    for (int m = 0; m < 4; ++m) {
        const int mbase = m0 + m * 16 + ((lane >> 4) << 3);
        #pragma unroll
        for (int r = 0; r < 8; ++r)
            S[(size_t)(mbase + r) * MTOT + ncol] = acc[m][r];
    }
}

// =====================================================================
// Kernel 2: per-row loss. One 512-thread block per row.
// =====================================================================
#define NT 512

__global__ __launch_bounds__(NT)
void capmem_rowloss(const float* __restrict__ S,
                    const int* __restrict__ tgts,
                    const int* __restrict__ camsArr,
                    float* __restrict__ partials)
{
    __shared__ float wv[MTOT];          // 64KB working copy of the row
    __shared__ float rv[NT];
    __shared__ int   ri[NT];
    __shared__ float camv[NCAM];
    __shared__ int   cami[NCAM];
    __shared__ float topv[KTOP];
    __shared__ int   topi[KTOP];

    const int b   = blockIdx.x;
    const int tid = threadIdx.x;
    const int cam = camsArr[b];
    const int tgt = tgts[b];
    const int g   = cam * CC + tgt;
    const float* row = S + (size_t)b * MTOT;

    for (int i = tid; i < MTOT; i += NT) wv[i] = row[i];
    __syncthreads();

    const float sg = wv[g];             // own-proxy similarity (pre-destruction)

    // ---------- CE over own camera block ----------
    float lm = -INFINITY;
    for (int j = tid; j < CC; j += NT) lm = fmaxf(lm, wv[cam * CC + j]);
    rv[tid] = lm;
    for (int s = NT / 2; s > 0; s >>= 1) {
        __syncthreads();
        if (tid < s) rv[tid] = fmaxf(rv[tid], rv[tid + s]);
    }
    __syncthreads();
    const float cem = rv[0];
    __syncthreads();
    float ssum = 0.f;
    for (int j = tid; j < CC; j += NT)
        ssum += __expf((wv[cam * CC + j] - cem) / BETA);
    rv[tid] = ssum;
    for (int s = NT / 2; s > 0; s >>= 1) {
        __syncthreads();
        if (tid < s) rv[tid] += rv[tid + s];
    }
    __syncthreads();
    const float ce = cem / BETA + __logf(rv[0]) - sg / BETA;
    __syncthreads();

    // ---------- per-camera argmax (8 pairs) ----------
    for (int c = 0; c < NCAM; ++c) {
        float bv = -INFINITY; int bi = 0;
        for (int j = tid; j < CC; j += NT) {
            float v = wv[c * CC + j];
            if (v > bv) { bv = v; bi = j; }     // ascending j => lowest idx on tie
        }
        rv[tid] = bv; ri[tid] = bi;
        for (int s = NT / 2; s > 0; s >>= 1) {
            __syncthreads();
            if (tid < s) {
                float v1 = rv[tid + s]; int i1 = ri[tid + s];
                if (v1 > rv[tid] || (v1 == rv[tid] && i1 < ri[tid])) {
                    rv[tid] = v1; ri[tid] = i1;
                }
            }
        }
        __syncthreads();
        if (tid == 0) { camv[c] = rv[0]; cami[c] = c * CC + ri[0]; }
        __syncthreads();
    }

    // ---------- top-KTOP (value,index) of whole row, destructive ----------
    for (int k = 0; k < KTOP; ++k) {
        float bv = -INFINITY; int bi = 0;
        for (int i = tid; i < MTOT; i += NT) {
            float v = wv[i];
            if (v > bv) { bv = v; bi = i; }
        }
        rv[tid] = bv; ri[tid] = bi;
        for (int s = NT / 2; s > 0; s >>= 1) {
            __syncthreads();
            if (tid < s) {
                float v1 = rv[tid + s]; int i1 = ri[tid + s];
                if (v1 > rv[tid] || (v1 == rv[tid] && i1 < ri[tid])) {
                    rv[tid] = v1; ri[tid] = i1;
                }
            }
        }
        __syncthreads();
        if (tid == 0) {
            topv[k] = rv[0]; topi[k] = ri[0];
            wv[ri[0]] = -INFINITY;
        }
        __syncthreads();
    }

    // ---------- final assembly on thread 0 ----------
    if (tid == 0) {
        // per-cam weight 1/count
        int cnt = 0;
        for (int i = 0; i < BSZ; ++i) if (camsArr[i] == cam) ++cnt;
        const float w = 1.0f / (float)cnt;

        // top-3 of the 8 per-cam maxima (ties -> lower cam index)
        float cv[NCAM]; int ci[NCAM]; bool used[NCAM];
        for (int c = 0; c < NCAM; ++c) { cv[c] = camv[c]; ci[c] = cami[c]; used[c] = false; }
        float tv[TOPC]; int ti[TOPC];
        for (int k = 0; k < TOPC; ++k) {
            int best = -1; float bvv = -INFINITY;
            for (int c = 0; c < NCAM; ++c)
                if (!used[c] && cv[c] > bvv) { bvv = cv[c]; best = c; }
            used[best] = true; tv[k] = cv[best]; ti[k] = ci[best];
        }

        // asso: target + top-50 hardest negatives (own proxy excluded)
        float av[1 + 50];
        av[0] = sg;
        int ac = 1;
        for (int k = 0; k < KTOP && ac < 51; ++k)
            if (topi[k] != g) av[ac++] = topv[k];
        float am = av[0];
        for (int i = 1; i < 51; ++i) am = fmaxf(am, av[i]);
        float as = 0.f;
        for (int i = 0; i < 51; ++i) as += __expf((av[i] - am) / BETA);
        const float asso = am / BETA + __logf(as) - av[0] / BETA;

        // online: 3 forced per-cam tops + top-5 of the rest
        float ov[TOPC + EXTRA];
        ov[0] = tv[0]; ov[1] = tv[1]; ov[2] = tv[2];
        int oc = TOPC;
        for (int k = 0; k < KTOP && oc < TOPC + EXTRA; ++k) {
            int idx = topi[k];
            if (idx != ti[0] && idx != ti[1] && idx != ti[2]) ov[oc++] = topv[k];
        }
        float om = ov[0];
        for (int i = 1; i < TOPC + EXTRA; ++i) om = fmaxf(om, ov[i]);
        float os = 0.f;
        for (int i = 0; i < TOPC + EXTRA; ++i) os += __expf((ov[i] - om) / BETA);
        const float olse = om / BETA + __logf(os);
        const float onl  = olse - (ov[0] + ov[1] + ov[2]) / (3.0f * BETA);

        partials[b] = (0.6f * ce + 0.7f * asso + 0.7f * onl) * w;
    }
}

// =====================================================================
// Kernel 3: deterministic tree-sum of 256 partials -> d_out[0]
// =====================================================================
__global__ __launch_bounds__(256)
void capmem_finalsum(const float* __restrict__ partials, float* __restrict__ out)
{
    __shared__ float r[256];
    const int tid = threadIdx.x;
    r[tid] = partials[tid];
    for (int s = 128; s > 0; s >>= 1) {
        __syncthreads();
        if (tid < s) r[tid] += r[tid + s];
    }
    __syncthreads();
    if (tid == 0) out[0] = r[0];
}

// =====================================================================
extern "C" void kernel_launch(void* const* d_in, const int* in_sizes, int n_in,
                              void* d_out, int out_size, void* d_ws, size_t ws_size,
                              hipStream_t stream)
{
    const float* features = (const float*)d_in[0];
    // d_in[1] = global_features (unused by the reference)
    const float* mem      = (const float*)d_in[2];   // [8][2048][2048] -> [16384][2048]
    const int*   tgts     = (const int*)d_in[3];
    const int*   cams     = (const int*)d_in[4];
    float*       out      = (float*)d_out;

    float* sims     = (float*)d_ws;                          // 256*16384 f32 = 16 MB
    float* partials = sims + (size_t)BSZ * MTOT;             // 256 f32

    capmem_gemm<<<dim3(MTOT / 128, BSZ / MT), 256, 0, stream>>>(features, mem, sims);
    capmem_rowloss<<<BSZ, NT, 0, stream>>>(sims, tgts, cams, partials);
    capmem_finalsum<<<1, 256, 0, stream>>>(partials, out);
}